// ILWT53_2D_17858474016990
// MI455X (gfx1250) — compile-verified
//
#include <hip/hip_runtime.h>

// Forward 2D LeGall 5/3 lifting wavelet, fused single pass.
// Input  x  : (8, 32, 512, 512) fp32
// Output out: (8, 128, 256, 256) fp32  = concat[LL, LH, HL, HH] on axis 1.
//
// Bandwidth-bound: 512 MB min traffic -> ~22us @ 23.3 TB/s. Strategy:
//  - one fused pass through LDS tiles (no global intermediate)
//  - CDNA5 async-to-LDS loads (global_load_async_to_lds_b64/b128) with the
//    reflect-mirror folded into per-lane addresses; s_wait_asynccnt sync
//  - non-temporal 16B subband stores (output is write-once; keep input
//    halo lines resident in the 192MB L2 instead)

typedef __attribute__((ext_vector_type(4))) float v4f;  // native vector: OK for NT builtin

#define TR 32               // output rows per tile (per subband)
#define TC 32               // output cols per tile (per subband)
#define ROWS (2*TR + 8)     // 72 staged input/intermediate rows (halo 4+4)
#define PAIRS (TC + 4)      // 36 (even,odd) column pairs staged (halo 2+2)
#define XPW (2*PAIRS)       // 72 floats per staged input row
#define DLW (TC + 3)        // 35: 34 d-values padded to odd stride
#define SLW (TC + 1)        // 33: 32 s-values padded to odd stride
#define CHUNKS (PAIRS/2)    // 18 x 16B chunks per staged row

// Half-sample mirror used by the reference's reflect_pad on even/odd/d streams.
__device__ __forceinline__ int mirror_half(int m, int nh) {
    m = (m < 0) ? -m : m;
    m = (m >= nh) ? (2 * (nh - 1) - m) : m;
    return m;
}

__global__ __launch_bounds__(256)
void lwt53_fwd2d_kernel(const float* __restrict__ x, float* __restrict__ out) {
    __shared__ __align__(16) float xp[ROWS][XPW];  // staged input (even,odd interleaved)
    __shared__ __align__(16) float dl[ROWS][DLW];  // row-transform d, i_loc in [-1,32]
    __shared__ __align__(16) float sl[ROWS][SLW];  // row-transform s, i_loc in [0,32)

    const int bx  = blockIdx.x;
    const int ti  = bx & 7;          // column tile
    const int tj  = (bx >> 3) & 7;   // row tile
    const int ch  = (bx >> 6) & 31;
    const int bb  = bx >> 11;
    const int I0  = ti * TC;         // first output column index (0..255)
    const int J0  = tj * TR;         // first output row index (0..255)
    const int tid = threadIdx.x;

    const float* slice = x + (size_t)(bb * 32 + ch) * (512 * 512);

    // ---- Phase 1: async gather of the 72x72 input tile into LDS.
    // Row r and column-pair m indices are mirrored at the half-sample level,
    // which exactly reproduces the reference's reflect_pad on even/odd/d.
    // The row mirror only reselects the source row (contiguity preserved), so
    // column-interior tiles can use 16B async loads; edge tiles gather 8B pairs.
    const int R0 = 2 * J0 - 4;   // first staged input row (may be negative)
    const int M0 = I0 - 2;       // first staged column pair (may be negative)
    if (ti != 0 && ti != 7) {
        // Interior in the column direction: all 36 pairs are contiguous.
        for (int idx = tid; idx < ROWS * CHUNKS; idx += 256) {
            const int rloc = idx / CHUNKS;
            const int c4   = idx - rloc * CHUNKS;       // 16B chunk = 2 pairs
            const int r    = R0 + rloc;
            const int p    = r & 1;
            const int rg   = 2 * mirror_half(r >> 1, 256) + p;
            const int mg   = M0 + 2 * c4;               // in range, no mirror
            const float* gsrc = slice + (size_t)rg * 512 + 2 * mg;   // 16B aligned
            const unsigned lofs = (unsigned)(uintptr_t)&xp[rloc][4 * c4];
            asm volatile("global_load_async_to_lds_b128 %0, %1, off"
                         :: "v"(lofs), "v"(gsrc) : "memory");
        }
    } else {
        // Image left/right edge: per-pair gather with mirrored column index.
        for (int idx = tid; idx < ROWS * PAIRS; idx += 256) {
            const int rloc = idx / PAIRS;
            const int mloc = idx - rloc * PAIRS;
            const int r    = R0 + rloc;
            const int p    = r & 1;                     // parity survives mirror
            const int rg   = 2 * mirror_half(r >> 1, 256) + p;
            const int mg   = mirror_half(M0 + mloc, 256);
            const float* gsrc = slice + (size_t)rg * 512 + 2 * mg;   // 8B aligned
            const unsigned lofs = (unsigned)(uintptr_t)&xp[rloc][2 * mloc];
            asm volatile("global_load_async_to_lds_b64 %0, %1, off"
                         :: "v"(lofs), "v"(gsrc) : "memory");
        }
    }
    asm volatile("s_wait_asynccnt 0" ::: "memory");
    __syncthreads();

    // ---- Phase 2: row predict. dl[r][k] holds d at i_loc = k-1, k in [0,34).
    // d(i) = odd(i) - 0.5*(E(i-1) + E(i+1));  E(i)=xp[.][2*(i+2)], odd(i)=xp[.][2*(i+2)+1]
    for (int idx = tid; idx < ROWS * (TC + 2); idx += 256) {
        const int r = idx / (TC + 2);
        const int k = idx - r * (TC + 2);
        dl[r][k] = xp[r][2 * k + 3] - 0.5f * (xp[r][2 * k] + xp[r][2 * k + 4]);
    }
    __syncthreads();

    // ---- Phase 3: row update. s(i) = E(i) + 0.25*(d(i-1) + d(i+1)), i in [0,32)
    for (int idx = tid; idx < ROWS * TC; idx += 256) {
        const int r = idx / TC;
        const int i = idx - r * TC;
        sl[r][i] = xp[r][2 * i + 4] + 0.25f * (dl[r][i] + dl[r][i + 2]);
    }
    __syncthreads();

    // ---- Phase 4: column lifting on sl (-> LL,LH) and dl (-> HL,HH), store.
    // Staged row rloc corresponds to global row R0 + rloc, so for local output
    // row j: E2(j) = sl[2j+4][.], odd2(j) = sl[2j+5][.].
    const int j  = tid >> 3;          // 0..31
    const int i0 = (tid & 7) * 4;     // 0,4,...,28
    v4f vLL, vLH, vHL, vHH;
    #pragma unroll
    for (int di = 0; di < 4; ++di) {
        const int i = i0 + di;
        // s-column lifting
        const float dm1 = sl[2*j + 3][i] - 0.5f * (sl[2*j    ][i] + sl[2*j + 4][i]);
        const float dc0 = sl[2*j + 5][i] - 0.5f * (sl[2*j + 2][i] + sl[2*j + 6][i]);
        const float dp1 = sl[2*j + 7][i] - 0.5f * (sl[2*j + 4][i] + sl[2*j + 8][i]);
        vLL[di] = sl[2*j + 4][i] + 0.25f * (dm1 + dp1);
        vLH[di] = dc0;
        // d-column lifting (output column i lives at dl index i+1)
        const int ic = i + 1;
        const float hm1 = dl[2*j + 3][ic] - 0.5f * (dl[2*j    ][ic] + dl[2*j + 4][ic]);
        const float hc0 = dl[2*j + 5][ic] - 0.5f * (dl[2*j + 2][ic] + dl[2*j + 6][ic]);
        const float hp1 = dl[2*j + 7][ic] - 0.5f * (dl[2*j + 4][ic] + dl[2*j + 8][ic]);
        vHL[di] = dl[2*j + 4][ic] + 0.25f * (hm1 + hp1);
        vHH[di] = hc0;
    }

    float* obase = out + (size_t)bb * (128 * 256 * 256) + (size_t)ch * (256 * 256);
    const size_t bandStride = (size_t)32 * 256 * 256;
    const size_t o = (size_t)(J0 + j) * 256 + (size_t)(I0 + i0);
    // Output is write-once/streaming: non-temporal so it doesn't evict the
    // input halo working set from L2.
    __builtin_nontemporal_store(vLL, reinterpret_cast<v4f*>(obase + 0 * bandStride + o));
    __builtin_nontemporal_store(vLH, reinterpret_cast<v4f*>(obase + 1 * bandStride + o));
    __builtin_nontemporal_store(vHL, reinterpret_cast<v4f*>(obase + 2 * bandStride + o));
    __builtin_nontemporal_store(vHH, reinterpret_cast<v4f*>(obase + 3 * bandStride + o));
}

extern "C" void kernel_launch(void* const* d_in, const int* in_sizes, int n_in,
                              void* d_out, int out_size, void* d_ws, size_t ws_size,
                              hipStream_t stream) {
    const float* x = (const float*)d_in[0];
    float* outp = (float*)d_out;
    // 8 batch * 32 ch * (8x8) tiles = 16384 blocks, 256 threads (8 wave32s).
    lwt53_fwd2d_kernel<<<dim3(16384), dim3(256), 0, stream>>>(x, outp);
}